// Embed_87170656239793
// MI455X (gfx1250) — compile-verified
//
#include <hip/hip_runtime.h>
#include <math.h>

#if __has_include(<hip/amd_detail/amd_gfx1250_TDM.h>)
#define TDM_6ARG 1
#endif

#define BB   64
#define NN   512
#define KNB  16
#define DD   128
#define MM   (BB * NN)            // 32768 rows

#define A_STRIDE  132             // 128 + 4 pad dwords/row (TDM pad) -> conflict-free frags
#define A_FLOATS  (16 * A_STRIDE) // 2112 floats = 8448 B
#define B_FLOATS  (32 * DD * 2 * 2) // 16384 floats = 64 KB (packed weight)
#define SM_FLOATS (A_FLOATS + B_FLOATS)

typedef __attribute__((ext_vector_type(2))) float        v2f;
typedef __attribute__((ext_vector_type(8))) float        v8f;
typedef __attribute__((ext_vector_type(4))) unsigned int v4u;
typedef __attribute__((ext_vector_type(8))) int          v8i;
typedef __attribute__((ext_vector_type(4))) int          v4i;

// ---------------------------------------------------------------------------
// TDM 2-D load: global (td0 x td1 tensor, row stride stride0 elems, 4B elems)
// tile (tl0 x tl1) -> LDS byte offset lds_off, optional LDS row padding.
// D# layout per CDNA5 ISA 8.3/8.4 (group0: count/lds/global/type, group1:
// data_size, dims, tile, stride, pad). Groups 2/3 zero (<=2D tensor).
// ---------------------------------------------------------------------------
__device__ __forceinline__ void tdm_load_2d(unsigned lds_off, const void* gsrc,
                                            unsigned td0, unsigned td1,
                                            unsigned tl0, unsigned tl1,
                                            unsigned stride0,
                                            unsigned pad_en, unsigned pad_int,
                                            unsigned pad_amt)
{
    unsigned long long ga = (unsigned long long)gsrc;
    v4u g0;
    g0[0] = 1u;                                     // count=1 (valid user D#)
    g0[1] = lds_off;                                // lds_addr (bytes)
    g0[2] = (unsigned)ga;                           // global_addr[31:0]
    g0[3] = (unsigned)((ga >> 32) & 0x01FFFFFFu)    // global_addr[56:32]
          | 0x80000000u;                            // type=2 ("image")
    v8i g1;
    g1[0] = (int)((2u << 16) | (pad_en << 20) | (pad_int << 22) | (pad_amt << 25));
    g1[1] = (int)((td0 & 0xFFFFu) << 16);                                   // dim0[15:0]
    g1[2] = (int)(((td0 >> 16) & 0xFFFFu) | ((td1 & 0xFFFFu) << 16));       // dim0 hi | dim1 lo
    g1[3] = (int)(((td1 >> 16) & 0xFFFFu) | ((tl0 & 0xFFFFu) << 16));       // dim1 hi | tile0
    g1[4] = (int)(tl1 & 0xFFFFu);                                           // tile1 (tile2=0)
    g1[5] = (int)stride0;                                                   // dim0_stride lo32
    g1[6] = 0;                                                              // stride hi / dim1_stride
    g1[7] = 0;
    v4i gz = {0, 0, 0, 0};
#ifdef TDM_6ARG
    v8i gz8 = {0, 0, 0, 0, 0, 0, 0, 0};
    __builtin_amdgcn_tensor_load_to_lds(g0, g1, gz, gz, gz8, 0);
#else
    __builtin_amdgcn_tensor_load_to_lds(g0, g1, gz, gz, 0);
#endif
}

// ---------------------------------------------------------------------------
// 16x16 tile, K=128: 32 x V_WMMA_F32_16X16X4_F32 reading LDS only.
// A frag: lane(rl,half) -> sA[rl*132 + half*2 + kk*4 .. +1]  (banks 4rl+2half)
// B frag: packed order Bp[kk][n][half][2] -> one contiguous b64 per lane.
// ---------------------------------------------------------------------------
__device__ __forceinline__ v8f wmma_tile_lds(const float* __restrict__ sA,
                                             const float* __restrict__ sB,
                                             int rl, int half, int col0, v8f acc)
{
    const float* aF = sA + rl * A_STRIDE + half * 2;
    const float* bF = sB + ((col0 + rl) * 2 + half) * 2;
#pragma unroll
    for (int kk = 0; kk < DD / 4; ++kk) {
        v2f a, b;
        a.x = aF[kk * 4 + 0];
        a.y = aF[kk * 4 + 1];
        b.x = bF[kk * 512 + 0];
        b.y = bF[kk * 512 + 1];
        acc = __builtin_amdgcn_wmma_f32_16x16x4_f32(false, a, false, b,
                                                    (short)0, acc, false, false);
    }
    return acc;
}

// ---------------------------------------------------------------------------
// Repack weight W[128][128] into B-fragment order:
//   Bp[((kk*128 + n)*2 + half)*2 + j] = W[(kk*4 + half*2 + j)*128 + n]
// ---------------------------------------------------------------------------
__global__ void k_pack_w(const float* __restrict__ W, float* __restrict__ Bp)
{
    int t    = blockIdx.x * 256 + threadIdx.x;   // 0..16383
    int j    = t & 1;
    int half = (t >> 1) & 1;
    int n    = (t >> 2) & 127;
    int kk   = t >> 9;
    Bp[t] = W[(size_t)(kk * 4 + half * 2 + j) * DD + n];
}

// ---------------------------------------------------------------------------
// Gather-mean over KNB neighbors (indices block-uniform -> scalar loads).
// ---------------------------------------------------------------------------
__global__ void k_agg_mean(const float* __restrict__ src, const int* __restrict__ idx,
                           float* __restrict__ out)
{
    int row = blockIdx.x;            // b*NN + n
    int b   = row >> 9;              // NN == 512
    int d   = threadIdx.x;           // 0..127
    const int*   ip = idx + (size_t)row * KNB;
    const float* sb = src + (size_t)b * NN * DD;
    float s = 0.f;
#pragma unroll
    for (int k = 0; k < KNB; ++k)
        s += sb[(size_t)ip[k] * DD + d];
    out[(size_t)row * DD + d] = s * (1.0f / KNB);
}

// ---------------------------------------------------------------------------
// base = A0@W0 + A1@W1 ; E1 = tanh(base).  One 16-row stripe per block,
// 8 waves = 8 n-tiles.  A stripe + packed W staged by TDM into LDS.
// ---------------------------------------------------------------------------
__global__ void __launch_bounds__(256)
k_gemm2_base_tanh(const float* __restrict__ A0, const float* __restrict__ Bp0,
                  const float* __restrict__ A1, const float* __restrict__ Bp1,
                  float* __restrict__ base, float* __restrict__ E1)
{
    __shared__ float smem[SM_FLOATS];
    int lane = threadIdx.x & 31, wid = threadIdx.x >> 5;
    int rl   = lane & 15,        half = lane >> 4;
    int row0 = blockIdx.x * 16,  col0 = wid * 16;
    unsigned ldsA = (unsigned)(unsigned long long)(void*)&smem[0];
    unsigned ldsB = (unsigned)(unsigned long long)(void*)&smem[A_FLOATS];

    const float* Aarr[2] = {A0, A1};
    const float* Barr[2] = {Bp0, Bp1};

    v8f acc = {};
#pragma unroll 1
    for (int p = 0; p < 2; ++p) {
        __syncthreads();                       // LDS reuse between passes
        if (wid == 0) {
            tdm_load_2d(ldsA, Aarr[p] + (size_t)row0 * DD,
                        DD, 16, DD, 16, DD, /*pad*/1, /*128dw*/6, /*4dw*/3);
            tdm_load_2d(ldsB, Barr[p],
                        B_FLOATS, 1, B_FLOATS, 1, B_FLOATS, 0, 0, 0);
            __builtin_amdgcn_s_wait_tensorcnt(0);
        }
        __syncthreads();
        asm volatile("" ::: "memory");
        acc = wmma_tile_lds(smem, smem + A_FLOATS, rl, half, col0, acc);
    }

    int col = col0 + rl, rb = row0 + half * 8;
#pragma unroll
    for (int r = 0; r < 8; ++r) {
        float v = acc[r];
        size_t o = (size_t)(rb + r) * DD + col;
        base[o] = v;
        E1[o]   = tanhf(v);
    }
}

// ---------------------------------------------------------------------------
// E2 = tanh(base + AG@W2)
// ---------------------------------------------------------------------------
__global__ void __launch_bounds__(256)
k_gemm_addbase_tanh(const float* __restrict__ AG, const float* __restrict__ BpW,
                    const float* __restrict__ base, float* __restrict__ E2)
{
    __shared__ float smem[SM_FLOATS];
    int lane = threadIdx.x & 31, wid = threadIdx.x >> 5;
    int rl   = lane & 15,        half = lane >> 4;
    int row0 = blockIdx.x * 16,  col0 = wid * 16;
    unsigned ldsA = (unsigned)(unsigned long long)(void*)&smem[0];
    unsigned ldsB = (unsigned)(unsigned long long)(void*)&smem[A_FLOATS];

    if (wid == 0) {
        tdm_load_2d(ldsA, AG + (size_t)row0 * DD,
                    DD, 16, DD, 16, DD, 1, 6, 3);
        tdm_load_2d(ldsB, BpW,
                    B_FLOATS, 1, B_FLOATS, 1, B_FLOATS, 0, 0, 0);
        __builtin_amdgcn_s_wait_tensorcnt(0);
    }
    __syncthreads();
    asm volatile("" ::: "memory");

    v8f acc = {};
    acc = wmma_tile_lds(smem, smem + A_FLOATS, rl, half, col0, acc);

    int col = col0 + rl, rb = row0 + half * 8;
#pragma unroll
    for (int r = 0; r < 8; ++r) {
        size_t o = (size_t)(rb + r) * DD + col;
        E2[o] = tanhf(acc[r] + base[o]);
    }
}

// ---------------------------------------------------------------------------
// pooled[b,:] = sum_n E2[b,n,:]*mask[b,n] ; out = pooled @ W2
// ---------------------------------------------------------------------------
__global__ void k_pool_out(const float* __restrict__ E2, const float* __restrict__ mask,
                           const float* __restrict__ W2, float* __restrict__ out)
{
    int b = blockIdx.x;
    int e = threadIdx.x;             // 128 threads
    __shared__ float pooled[DD];
    const float* Eb = E2   + (size_t)b * NN * DD;
    const float* mb = mask + (size_t)b * NN;
    float s = 0.f;
    for (int n = 0; n < NN; ++n)
        s += Eb[(size_t)n * DD + e] * mb[n];   // mb[n] uniform -> scalar load
    pooled[e] = s;
    __syncthreads();
    float o = 0.f;
#pragma unroll 8
    for (int k = 0; k < DD; ++k)
        o += pooled[k] * W2[(size_t)k * DD + e];
    out[(size_t)b * DD + e] = o;
}

extern "C" void kernel_launch(void* const* d_in, const int* in_sizes, int n_in,
                              void* d_out, int out_size, void* d_ws, size_t ws_size,
                              hipStream_t stream)
{
    (void)in_sizes; (void)n_in; (void)out_size; (void)ws_size;

    const float* word_embs = (const float*)d_in[0];
    const int*   neibors   = (const int*)  d_in[1];
    const float* lib_embs  = (const float*)d_in[2];
    const int*   neib_lib  = (const int*)  d_in[3];
    const float* mask      = (const float*)d_in[4];
    const float* weight    = (const float*)d_in[5];
    const float* weight2   = (const float*)d_in[6];
    const float* enc_w1    = (const float*)d_in[7];
    const float* enc_w2    = (const float*)d_in[8];
    float* out = (float*)d_out;

    const size_t big = (size_t)MM * DD;        // 16 MB each
    float* buf0 = (float*)d_ws;                // agg_lib, then agg(E1)
    float* buf1 = buf0 + big;                  // base
    float* buf2 = buf1 + big;                  // E1, then E2
    float* bpW  = buf2 + big;                  // packed weight
    float* bpW1 = bpW  + B_FLOATS;             // packed enc_w1
    float* bpW2 = bpW1 + B_FLOATS;             // packed enc_w2

    // 0) repack the three weight matrices into B-fragment order
    k_pack_w<<<64, 256, 0, stream>>>(weight, bpW);
    k_pack_w<<<64, 256, 0, stream>>>(enc_w1, bpW1);
    k_pack_w<<<64, 256, 0, stream>>>(enc_w2, bpW2);

    dim3 aggGrid(MM), aggBlk(DD);
    dim3 gGrid(MM / 16), gBlk(256);            // 2048 blocks, 8 waves each

    // 1) agg_lib_raw = mean-gather(lib_embs)
    k_agg_mean<<<aggGrid, aggBlk, 0, stream>>>(lib_embs, neib_lib, buf0);
    // 2) base = agg_lib_raw@weight + word@enc_w1 ; E1 = tanh(base)
    k_gemm2_base_tanh<<<gGrid, gBlk, 0, stream>>>(buf0, bpW, word_embs, bpW1, buf1, buf2);
    // 3) AG = mean-gather(E1)
    k_agg_mean<<<aggGrid, aggBlk, 0, stream>>>(buf2, neibors, buf0);
    // 4) E2 = tanh(base + AG@enc_w2)   (overwrites E1, which is dead)
    k_gemm_addbase_tanh<<<gGrid, gBlk, 0, stream>>>(buf0, bpW2, buf1, buf2);
    // 5) masked pool + final linear
    k_pool_out<<<BB, DD, 0, stream>>>(buf2, mask, weight2, out);
}